// PoseLoss_3427383902463
// MI455X (gfx1250) — compile-verified
//
#include <hip/hip_runtime.h>

// Problem constants (match reference)
#define BB 16
#define SS 2048
#define PP 4096
#define BLOCKS_PER_BATCH 8   // 2048 slices / 256 threads
#define TPB 256

#if defined(__has_builtin)
#if __has_builtin(__builtin_amdgcn_global_load_async_to_lds_b32)
#define HAVE_ASYNC_LDS 1
#endif
#if __has_builtin(__builtin_amdgcn_s_wait_asynccnt)
#define HAVE_WAIT_ASYNC 1
#endif
#endif

#ifdef HAVE_ASYNC_LDS
#define ASYNC_COPY_B32(GP, LP)                                              \
  __builtin_amdgcn_global_load_async_to_lds_b32(                            \
      (__attribute__((address_space(1))) int*)(GP),                         \
      (__attribute__((address_space(3))) int*)(LP), 0, 0)
#endif

__device__ __forceinline__ void wait_async_zero() {
#ifdef HAVE_ASYNC_LDS
#ifdef HAVE_WAIT_ASYNC
  __builtin_amdgcn_s_wait_asynccnt(0);
#else
  asm volatile("s_wait_asynccnt 0" ::: "memory");
#endif
#endif
}

// One block handles 256 consecutive slices of one batch.
// grid = BB * BLOCKS_PER_BATCH, block = TPB.
__global__ void pose_loss_main(const float* __restrict__ pred,
                               const float* __restrict__ gt,
                               float* __restrict__ partials) {
  __shared__ float lt[PP];        // 16 KB: gt times for this batch
  __shared__ float red_t[TPB / 32];
  __shared__ float red_i[TPB / 32];

  const int b     = blockIdx.x / BLOCKS_PER_BATCH;
  const int chunk = blockIdx.x % BLOCKS_PER_BATCH;
  const int tid   = threadIdx.x;

  const float* __restrict__ gtb = gt + (size_t)b * PP * 8;

  // ---- Stage gt_times[b, :, 0] (stride-8 gather) into contiguous LDS ----
  // Static trip count -> fully unrolls into 16 back-to-back async issues.
#ifdef HAVE_ASYNC_LDS
#pragma unroll
  for (int k = 0; k < PP / TPB; ++k) {
    const int i = k * TPB + tid;
    ASYNC_COPY_B32(gtb + (size_t)i * 8, &lt[i]);
  }
  wait_async_zero();
#else
#pragma unroll
  for (int k = 0; k < PP / TPB; ++k) {
    const int i = k * TPB + tid;
    lt[i] = gtb[(size_t)i * 8];
  }
#endif
  __syncthreads();

  // ---- Binary search: nearest sorted time to q, argmin-first semantics ----
  const int   s = chunk * TPB + tid;            // slice index in [0, SS)
  const float q = (float)s * 5.0f + 2.5f;

  int lo = 0, hi = PP;                          // lower_bound: first lt[i] >= q
  while (lo < hi) {
    int m = (lo + hi) >> 1;
    if (lt[m] < q) lo = m + 1; else hi = m;
  }
  int best;
  if (lo == 0) {
    best = 0;
  } else if (lo == PP) {
    best = PP - 1;
  } else {
    // lt[lo-1] < q <= lt[lo]; tie -> lower index (argmin picks first min)
    best = ((q - lt[lo - 1]) <= (lt[lo] - q)) ? (lo - 1) : lo;
  }
  // Duplicate timestamps: argmin returns the FIRST index of the minimal diff.
  while (best > 0 && lt[best - 1] == lt[best]) --best;

  // ---- Gather pose row (two b128s: 32B row, 32B-aligned) + loss terms ----
  const float4* __restrict__ growv =
      (const float4*)(gtb + (size_t)best * 8);
  const float4 glo = growv[0];   // [t, g1, g2, g3]
  const float4 ghi = growv[1];   // [g4, g5, g6, g7]

  const float* __restrict__ prow = pred + ((size_t)b * SS + s) * 7;
  float p0 = prow[0], p1 = prow[1], p2 = prow[2];
  float p3 = prow[3], p4 = prow[4], p5 = prow[5], p6 = prow[6];

  float d0 = p0 - glo.y, d1 = p1 - glo.z, d2 = p2 - glo.w;
  float tsq   = d0 * d0 + d1 * d1 + d2 * d2;
  float inner = fabsf(p3 * ghi.x + p4 * ghi.y + p5 * ghi.z + p6 * ghi.w);

  // ---- Deterministic block reduction: wave32 shuffle tree + LDS ----
  float vt = tsq, vi = inner;
#pragma unroll
  for (int off = 16; off > 0; off >>= 1) {
    vt += __shfl_xor(vt, off, 32);
    vi += __shfl_xor(vi, off, 32);
  }
  const int wave = tid >> 5, lane = tid & 31;
  if (lane == 0) { red_t[wave] = vt; red_i[wave] = vi; }
  __syncthreads();
  if (tid == 0) {
    float st = 0.0f, si = 0.0f;
#pragma unroll
    for (int w = 0; w < TPB / 32; ++w) { st += red_t[w]; si += red_i[w]; }
    partials[2 * blockIdx.x + 0] = st;
    partials[2 * blockIdx.x + 1] = si;
  }
}

// Single block of 128 threads sums the 128 per-block partials (deterministic tree).
__global__ void pose_loss_final(const float* __restrict__ partials,
                                float* __restrict__ out) {
  __shared__ float st[BB * BLOCKS_PER_BATCH];
  __shared__ float si[BB * BLOCKS_PER_BATCH];
  const int t = threadIdx.x;
  st[t] = partials[2 * t + 0];
  si[t] = partials[2 * t + 1];
  __syncthreads();
#pragma unroll
  for (int off = (BB * BLOCKS_PER_BATCH) / 2; off > 0; off >>= 1) {
    if (t < off) { st[t] += st[t + off]; si[t] += si[t + off]; }
    __syncthreads();
  }
  if (t == 0) {
    const float t_loss = st[0] / (float)(BB * SS * 3);
    const float r_loss = 1.0f - si[0] / (float)(BB * SS);
    out[0] = t_loss + r_loss;   // total
    out[1] = t_loss;
    out[2] = r_loss;
  }
}

extern "C" void kernel_launch(void* const* d_in, const int* in_sizes, int n_in,
                              void* d_out, int out_size, void* d_ws, size_t ws_size,
                              hipStream_t stream) {
  (void)in_sizes; (void)n_in; (void)out_size; (void)ws_size;
  const float* pred = (const float*)d_in[0];   // (16, 2048, 7) f32
  const float* gt   = (const float*)d_in[1];   // (16, 4096, 8) f32
  float* out        = (float*)d_out;           // 3 f32: total, t_loss, r_loss
  float* partials   = (float*)d_ws;            // 128 blocks * 2 floats

  pose_loss_main<<<BB * BLOCKS_PER_BATCH, TPB, 0, stream>>>(pred, gt, partials);
  pose_loss_final<<<1, BB * BLOCKS_PER_BATCH, 0, stream>>>(partials, out);
}